// FocalLossReid_7447473291778
// MI455X (gfx1250) — compile-verified
//
#include <hip/hip_runtime.h>

#define B_ 8
#define A_ 120000
#define M_ 128
#define BLK 256
#define NBLK ((A_ + BLK - 1) / BLK)

typedef __attribute__((ext_vector_type(2))) float v2f;
typedef __attribute__((ext_vector_type(8))) float v8f;
typedef __attribute__((ext_vector_type(4))) unsigned int u32x4;
typedef __attribute__((ext_vector_type(8))) int i32x8;
typedef __attribute__((ext_vector_type(4))) int i32x4;
typedef unsigned long long u64;

// ---------------------------------------------------------------------------
// Tensor Data Mover: 1-D f32 copy global -> LDS (CDNA5 TDM, D# per ISA ch.8)
// group0: count=1, lds_addr, global_addr, type=2
// group1: data_size=4B, tensor_dim0=nelem, tensor_dim1=1, tile_dim0=nelem,
//         tile_dim1=1, tensor_dim0_stride=nelem
// ---------------------------------------------------------------------------
__device__ __forceinline__ void tdm_load_f32_1d(const void* gsrc,
                                                unsigned lds_byte_off,
                                                unsigned nelem) {
  unsigned long long ga = (unsigned long long)gsrc;
  u32x4 g0;
  g0.x = 1u;                                           // count=1, user mode
  g0.y = lds_byte_off;                                 // lds_addr[63:32]
  g0.z = (unsigned)ga;                                 // global_addr[95:64]
  g0.w = ((unsigned)(ga >> 32) & 0x01FFFFFFu) | 0x80000000u; // addr hi | type=2
  i32x8 g1;
  g1[0] = 0x00020000;                                  // data_size=2 (4 bytes)
  g1[1] = (int)((nelem & 0xFFFFu) << 16);              // tensor_dim0 lo16
  g1[2] = (int)((nelem >> 16) | (1u << 16));           // tensor_dim0 hi | tensor_dim1=1
  g1[3] = (int)((nelem & 0xFFFFu) << 16);              // tile_dim0
  g1[4] = 1;                                           // tile_dim1=1, tile_dim2=0
  g1[5] = (int)nelem;                                  // tensor_dim0_stride lo32
  g1[6] = 0;
  g1[7] = 0;
  i32x4 z4 = {0, 0, 0, 0};
  i32x8 z8 = {0, 0, 0, 0, 0, 0, 0, 0};
  // 6-arg form (amdgpu-toolchain clang-23 / therock headers)
  __builtin_amdgcn_tensor_load_to_lds(g0, g1, z4, z4, z8, 0);
}

// ---------------------------------------------------------------------------
// Kernel 0: zero the atomically-updated workspace state (every call)
// ---------------------------------------------------------------------------
__global__ void k_init(u64* gkey1, u64* gkey2, float* accum) {
  for (int i = threadIdx.x; i < B_ * M_; i += BLK) {
    gkey1[i] = 0ull;
    gkey2[i] = 0ull;
  }
  if (threadIdx.x < B_ * 2) accum[threadIdx.x] = 0.0f;
}

// ---------------------------------------------------------------------------
// Kernel 1: per-anchor best GT (max/argmax over M) for both annotation sets,
// plus per-GT best anchor (argmax over A) via packed-u64 atomicMax.
// grid = (NBLK, B), block = 256
// ---------------------------------------------------------------------------
__global__ void __launch_bounds__(BLK) k_match(
    const float* __restrict__ anchors,                 // [A,4]
    const float* __restrict__ ann1,                    // [B,M,6]
    const float* __restrict__ ann2,                    // [B,M,6]
    float* __restrict__ bto1, float* __restrict__ bto2,
    int* __restrict__ bti1, int* __restrict__ bti2,
    u64* __restrict__ gkey1, u64* __restrict__ gkey2) {
  __shared__ float sann[2][M_][6];                     // 6 KB, filled by TDM
  __shared__ u64 skey[2][M_];                          // per-block argmax keys

  const int b = blockIdx.y;
  const int a = blockIdx.x * BLK + (int)threadIdx.x;
  const bool inb = (a < A_);

  for (int i = threadIdx.x; i < 2 * M_; i += BLK) skey[i >> 7][i & (M_ - 1)] = 0ull;

  if (threadIdx.x < 32) {                              // wave 0 issues the DMAs
    unsigned base = (unsigned)(unsigned long long)(void*)&sann[0][0][0];
    tdm_load_f32_1d(ann1 + (size_t)b * M_ * 6, base, M_ * 6);
    tdm_load_f32_1d(ann2 + (size_t)b * M_ * 6, base + M_ * 6 * 4, M_ * 6);
    __builtin_amdgcn_s_wait_tensorcnt(0);
  }
  __syncthreads();

  const int ca = inb ? a : 0;
  const float4 av = ((const float4*)anchors)[ca];
  const float aarea = (av.z - av.x) * (av.w - av.y);

  float best[2] = {-1.0f, -1.0f};
  int bidx[2] = {0, 0};

#pragma unroll 1
  for (int m = 0; m < M_; ++m) {
#pragma unroll
    for (int s = 0; s < 2; ++s) {
      float bx1 = sann[s][m][0], by1 = sann[s][m][1];
      float bx2 = sann[s][m][2], by2 = sann[s][m][3];
      float iw = fmaxf(fminf(av.z, bx2) - fmaxf(av.x, bx1), 0.0f);
      float ih = fmaxf(fminf(av.w, by2) - fmaxf(av.y, by1), 0.0f);
      float inter = iw * ih;
      float ua = fmaxf(aarea + (bx2 - bx1) * (by2 - by1) - inter, 1e-8f);
      float iou = inter / ua;
      if (iou > best[s]) { best[s] = iou; bidx[s] = m; }
      // pack (iou, ~anchor) -> max gives highest IoU, lowest anchor on ties
      u64 key = inb ? ((((u64)__float_as_uint(iou)) << 32) |
                       (u64)(0xFFFFFFFFu - (unsigned)a))
                    : 0ull;
      for (int off = 16; off > 0; off >>= 1) {
        u64 o = __shfl_xor(key, off, 32);
        key = (o > key) ? o : key;
      }
      if ((threadIdx.x & 31) == 0) atomicMax(&skey[s][m], key);
    }
  }

  if (inb) {
    size_t o = (size_t)b * A_ + a;
    bto1[o] = best[0]; bti1[o] = bidx[0];
    bto2[o] = best[1]; bti2[o] = bidx[1];
  }
  __syncthreads();
  for (int i = threadIdx.x; i < 2 * M_; i += BLK) {
    int s = i >> 7, m = i & (M_ - 1);
    atomicMax(&(s ? gkey2 : gkey1)[b * M_ + m], skey[s][m]);
  }
}

// ---------------------------------------------------------------------------
// Kernel 2: scatter override + targets + focal loss; block sums reduced with
// V_WMMA_F32_16X16X4_F32 (ones-matrix reduction), then atomicAdd per image.
// grid = (NBLK, B), block = 256
// ---------------------------------------------------------------------------
__global__ void __launch_bounds__(BLK) k_loss(
    const float* __restrict__ cls,                     // [B,A,1]
    const float* __restrict__ ann1,
    const float* __restrict__ ann2,
    const float* __restrict__ bto1, const float* __restrict__ bto2,
    const int* __restrict__ bti1, const int* __restrict__ bti2,
    const u64* __restrict__ gkey1, const u64* __restrict__ gkey2,
    float* __restrict__ accum) {
  __shared__ int sbp[2][M_];
  __shared__ float sid[2][M_];

  const int b = blockIdx.y;
  const int a = blockIdx.x * BLK + (int)threadIdx.x;
  const bool inb = (a < A_);

  for (int i = threadIdx.x; i < 2 * M_; i += BLK) {
    int s = i >> 7, m = i & (M_ - 1);
    u64 k = (s ? gkey2 : gkey1)[b * M_ + m];
    sbp[s][m] = (int)(0xFFFFFFFFu - (unsigned)(k & 0xFFFFFFFFu));
    sid[s][m] = (s ? ann2 : ann1)[((size_t)b * M_ + m) * 6 + 5];
  }
  __syncthreads();

  const size_t o = (size_t)b * A_ + (inb ? a : 0);
  float bto = bto1[o];
  float btn = bto2[o];
  int bi = bti1[o];
  int bj = bti2[o];
#pragma unroll 1
  for (int m = 0; m < M_; ++m) {                       // sequential scatter, last wins
    if (sbp[0][m] == a) { bto = 2.0f; bi = m; }
    if (sbp[1][m] == a) { btn = 2.0f; bj = m; }
  }

  float t = -1.0f;
  bool valid = ((bto >= 0.5f) && (btn >= 0.4f)) || ((btn >= 0.5f) && (bto >= 0.4f));
  bool same = (sid[0][bi] == sid[1][bj]);
  if (valid) t = same ? 1.0f : 0.0f;
  if ((bto < 0.4f) || (btn < 0.4f)) t = 0.0f;
  if ((bto < 0.5f) && (bto >= 0.4f) && (btn < 0.5f) && (btn >= 0.4f)) t = -1.0f;

  float c = cls[o];
  c = fminf(fmaxf(c, 1e-4f), 1.0f - 1e-4f);
  bool isp = (t == 1.0f);
  float af = isp ? 0.25f : 0.75f;
  float x = isp ? (1.0f - c) : c;
  float fw = af * x * x;                               // gamma = 2
  float bce = -(t * logf(c) + (1.0f - t) * logf(1.0f - c));
  float loss = (t != -1.0f) ? fw * bce : 0.0f;
  float npos = (t >= 1.0f) ? 1.0f : 0.0f;
  if (!inb) { loss = 0.0f; npos = 0.0f; }

  // Wave reduction via WMMA: A[m][k] has K=0 (lanes 0-15) / K=2 (lanes 16-31)
  // carrying per-lane values, K=1/3 = 0; B = all-ones -> D row m = s_m + s_{m+16}.
  v2f Al; Al.x = loss; Al.y = 0.0f;
  v2f Ap; Ap.x = npos; Ap.y = 0.0f;
  v2f Bo; Bo.x = 1.0f; Bo.y = 1.0f;
  v8f zc = {};
  v8f Dl = __builtin_amdgcn_wmma_f32_16x16x4_f32(false, Al, false, Bo, (short)0,
                                                 zc, false, false);
  v8f Dp = __builtin_amdgcn_wmma_f32_16x16x4_f32(false, Ap, false, Bo, (short)0,
                                                 zc, false, false);
  int lane = threadIdx.x & 31;
  if (lane == 0 || lane == 16) {                       // N=0 column: M=0..7 / 8..15
    float sl = 0.0f, sp = 0.0f;
#pragma unroll
    for (int i = 0; i < 8; ++i) { sl += Dl[i]; sp += Dp[i]; }
    atomicAdd(&accum[b * 2 + 0], sl);
    atomicAdd(&accum[b * 2 + 1], sp);
  }
}

// ---------------------------------------------------------------------------
// Kernel 3: per-image normalize + mean over B
// ---------------------------------------------------------------------------
__global__ void k_final(const float* __restrict__ accum, float* __restrict__ out) {
  float v = 0.0f;
  if (threadIdx.x < B_) {
    float s = accum[threadIdx.x * 2 + 0];
    float p = accum[threadIdx.x * 2 + 1];
    v = s / fmaxf(p, 1.0f);
  }
  for (int off = 4; off > 0; off >>= 1) v += __shfl_down(v, off, 32);
  if (threadIdx.x == 0) out[0] = v * (1.0f / (float)B_);
}

// ---------------------------------------------------------------------------
extern "C" void kernel_launch(void* const* d_in, const int* in_sizes, int n_in,
                              void* d_out, int out_size, void* d_ws, size_t ws_size,
                              hipStream_t stream) {
  const float* cls = (const float*)d_in[0];      // [B,A,1]
  const float* anchors = (const float*)d_in[1];  // [1,A,4]
  const float* ann1 = (const float*)d_in[2];     // [B,M,6]
  const float* ann2 = (const float*)d_in[3];     // [B,M,6]

  char* w = (char*)d_ws;
  u64* gkey1 = (u64*)w;  w += (size_t)B_ * M_ * sizeof(u64);
  u64* gkey2 = (u64*)w;  w += (size_t)B_ * M_ * sizeof(u64);
  float* accum = (float*)w; w += 256;            // B*2 floats, padded
  float* bto1 = (float*)w; w += (size_t)B_ * A_ * sizeof(float);
  float* bto2 = (float*)w; w += (size_t)B_ * A_ * sizeof(float);
  int* bti1 = (int*)w;     w += (size_t)B_ * A_ * sizeof(int);
  int* bti2 = (int*)w;     w += (size_t)B_ * A_ * sizeof(int);

  dim3 grid(NBLK, B_);
  k_init<<<1, BLK, 0, stream>>>(gkey1, gkey2, accum);
  k_match<<<grid, BLK, 0, stream>>>(anchors, ann1, ann2, bto1, bto2, bti1, bti2,
                                    gkey1, gkey2);
  k_loss<<<grid, BLK, 0, stream>>>(cls, ann1, ann2, bto1, bto2, bti1, bti2,
                                   gkey1, gkey2, accum);
  k_final<<<1, 32, 0, stream>>>(accum, (float*)d_out);
}